// QuantumConvFilter_65481071402541
// MI455X (gfx1250) — compile-verified
//
#include <hip/hip_runtime.h>
#include <math.h>

// QuantumConvFilter: out = 0.5 - 0.125 * sum_i t_i * C_i
//   C_i = cumprod(cos(params))_i, t_i = +/-1 from cumulative XOR of (patch>0).
// Memory-bound 2x2 stencil: staged via CDNA5 async global->LDS loads,
// computed 4 columns/thread with b128 LDS reads and b128 stores.

#define TR     32        // output rows per block tile
#define WIDTH  256       // input W (and row stride in floats)
#define HEIGHT 256
#define OW     255       // output width/height
#define NBC    (32 * 16) // B*C images

#if __has_builtin(__builtin_amdgcn_global_load_async_to_lds_b128)
#define HAVE_ASYNC_BUILTIN 1
#endif
#if __has_builtin(__builtin_amdgcn_s_wait_asynccnt)
#define HAVE_WAIT_BUILTIN 1
#endif

typedef int v4i __attribute__((vector_size(16)));
typedef __attribute__((address_space(1))) v4i g_v4i;   // global-AS int4
typedef __attribute__((address_space(3))) v4i l_v4i;   // LDS-AS int4
typedef __attribute__((address_space(3))) void las_void;

typedef float f4a __attribute__((ext_vector_type(4)));              // 16B-aligned
typedef float f4u __attribute__((ext_vector_type(4), aligned(4)));  // 4B-aligned store

__device__ __forceinline__ float qpix(float a0, float a1, float b0, float b1,
                                      float C0, float C1, float C2, float C3) {
    // cumulative parity of (elem > 0); reference: sign = (p>0) ? -1 : +1
    const bool q0 = a0 > 0.0f;
    const bool q1 = q0 ^ (a1 > 0.0f);
    const bool q2 = q1 ^ (b0 > 0.0f);
    const bool q3 = q2 ^ (b1 > 0.0f);
    const float s = (q0 ? -C0 : C0) + (q1 ? -C1 : C1) +
                    (q2 ? -C2 : C2) + (q3 ? -C3 : C3);
    return 0.5f - 0.125f * s;
}

__global__ __launch_bounds__(256) void
QuantumConvFilter_65481071402541_kernel(const float* __restrict__ x,
                                        const float* __restrict__ params,
                                        float* __restrict__ out) {
    __shared__ __align__(16) float tile[(TR + 1) * WIDTH];

    const int tid     = threadIdx.x;
    const int bc      = blockIdx.y;                 // image index in [0, 512)
    const int r0      = blockIdx.x * TR;            // first output row of tile
    const int rowsOut = min(TR, OW - r0);           // 32 (or 31 on last tile)
    const int rowsIn  = rowsOut + 1;                // input rows needed (<= 33)

    const float* img = x + (size_t)bc * (WIDTH * HEIGHT);

    // ---- Stage rowsIn full input rows into LDS via async 16B chunks ----
    // 64 chunks of 16B per 1KB row.
    const int chunks = rowsIn * (WIDTH / 4);
    for (int q = tid; q < chunks; q += 256) {
        const int row = q >> 6;           // q / 64
        const int off = (q & 63) << 2;    // float offset within row
        const float* gp = img + (size_t)(r0 + row) * WIDTH + off;
        float*       lp = &tile[row * WIDTH + off];
#if defined(HAVE_ASYNC_BUILTIN)
        __builtin_amdgcn_global_load_async_to_lds_b128(
            (g_v4i*)(void*)const_cast<float*>(gp),
            (l_v4i*)(void*)lp, /*offset=*/0, /*cpol=*/0);
#elif defined(__gfx1250__)
        {
            unsigned lds_off = (unsigned)(unsigned long long)((las_void*)(void*)lp);
            unsigned long long ga = (unsigned long long)gp;
            asm volatile("global_load_async_to_lds_b128 %0, %1, off"
                         :: "v"(lds_off), "v"(ga) : "memory");
        }
#else
        *(float4*)lp = *(const float4*)gp;
#endif
    }
#if defined(HAVE_ASYNC_BUILTIN) || defined(__gfx1250__)
#if defined(HAVE_WAIT_BUILTIN)
    __builtin_amdgcn_s_wait_asynccnt(0);
#else
    asm volatile("s_wait_asynccnt 0x0" ::: "memory");
#endif
#endif
    __syncthreads();

    // ---- Uniform cumulative cosine products (scalar loads, 4 cos/thread) ----
    const float C0 = __builtin_cosf(params[0]);
    const float C1 = C0 * __builtin_cosf(params[1]);
    const float C2 = C1 * __builtin_cosf(params[2]);
    const float C3 = C2 * __builtin_cosf(params[3]);

    float* oimg = out + (size_t)bc * (OW * OW);

    // ---- Compute from LDS: 4 columns/thread, b128 LDS reads + b128 stores ----
    // 64 groups per row: groups 0..62 are 4 aligned cols; group 63 = cols 252..254.
    const int ngroups = rowsOut * 64;
    for (int w = tid; w < ngroups; w += 256) {
        const int r = w >> 6;
        const int g = w & 63;
        const int obase = (r0 + r) * OW;
        const float* t0 = &tile[r * WIDTH];
        const float* t1 = t0 + WIDTH;
        if (g < 63) {
            const int c0 = g << 2;                    // 16B-aligned in LDS
            const f4a va = *(const f4a*)(t0 + c0);
            const float ea = t0[c0 + 4];
            const f4a vb = *(const f4a*)(t1 + c0);
            const float eb = t1[c0 + 4];
            f4u res;
            res.x = qpix(va.x, va.y, vb.x, vb.y, C0, C1, C2, C3);
            res.y = qpix(va.y, va.z, vb.y, vb.z, C0, C1, C2, C3);
            res.z = qpix(va.z, va.w, vb.z, vb.w, C0, C1, C2, C3);
            res.w = qpix(va.w, ea,   vb.w, eb,   C0, C1, C2, C3);
            *(f4u*)(oimg + obase + c0) = res;         // b128 store, 4B-aligned OK
        } else {
            #pragma unroll
            for (int c = 252; c < 255; ++c) {
                oimg[obase + c] = qpix(t0[c], t0[c + 1], t1[c], t1[c + 1],
                                       C0, C1, C2, C3);
            }
        }
    }
}

extern "C" void kernel_launch(void* const* d_in, const int* in_sizes, int n_in,
                              void* d_out, int out_size, void* d_ws, size_t ws_size,
                              hipStream_t stream) {
    const float* x      = (const float*)d_in[0];
    const float* params = (const float*)d_in[1];
    float*       out    = (float*)d_out;

    dim3 grid((OW + TR - 1) / TR, NBC);   // 8 x 512 blocks
    dim3 block(256);
    QuantumConvFilter_65481071402541_kernel<<<grid, block, 0, stream>>>(x, params, out);
}